// Gate_22935125360825
// MI455X (gfx1250) — compile-verified
//
#include <hip/hip_runtime.h>

// 26-qubit controlled 2-qubit gate: targets = bits 25,24; control = bit 23.
// Lifted to OUT = U16 @ X, U16 = 16x16 operator on bits {25,24,23,22},
// X = 16 rows (stride 2^22 elements) x 2^22 contiguous columns.
// Memory-bound: 512 MB @ 23.3 TB/s ~ 22 us floor; WMMA f32 does the math for free.

typedef __attribute__((ext_vector_type(2))) float v2f;
typedef __attribute__((ext_vector_type(8))) float v8f;

#define ROW_SHIFT 22                     // row stride = 2^22 elements (16 MiB)
#define TILES_PER_WAVE 4                 // 4 tiles x 16 cols = 64 columns per wave
#define WAVES_PER_BLOCK 8
#define THREADS (WAVES_PER_BLOCK * 32)   // 256 threads = 8 wave32

__global__ __launch_bounds__(THREADS) void ctrl_gate_wmma(
    const float* __restrict__ x, const float* __restrict__ m,
    float* __restrict__ out)
{
  const unsigned S    = 1u << ROW_SHIFT;
  const int lane = threadIdx.x & 31;
  const int hi   = lane >> 4;            // 0: lanes 0-15, 1: lanes 16-31
  const int mrow = lane & 15;            // output row M this lane owns in A; column N in D

  // ---- Build A = U16 in the ISA 16x4 f32 A-layout, one K-chunk per WMMA ----
  // U16[rp][r]: rp,r in 0..15 over bits (b25 b24 b23 b22).
  //   zero unless (rp&3)==(r&3)  (control bit + bit22 must match)
  //   control==0 -> identity on targets; control==1 -> M[tp][t], t=row>>2.
  // Chunk j layout: lanes 0-15 hold {K=4j+0, K=4j+1}; lanes 16-31 {K=4j+2, K=4j+3}.
  v2f A[4];
#pragma unroll
  for (int j = 0; j < 4; ++j) {
    const int c0 = 4 * j + 2 * hi;
#pragma unroll
    for (int p = 0; p < 2; ++p) {
      const int r  = c0 + p;             // input row index of U16
      const int tp = mrow >> 2;
      const int t  = r >> 2;
      const float mv  = m[tp * 4 + t];               // gate element (tiny, cached)
      const float idv = (tp == t) ? 1.0f : 0.0f;     // identity element
      float v = (((r >> 1) & 1) != 0) ? mv : idv;    // control bit of input row
      v = ((mrow & 3) == (r & 3)) ? v : 0.0f;        // block-diagonal mask
      if (p == 0) A[j].x = v; else A[j].y = v;
    }
  }

  // ---- Stream the state: each wave handles 64 columns (4 WMMA tiles) ----
  const unsigned wave = blockIdx.x * WAVES_PER_BLOCK + (threadIdx.x >> 5);
  unsigned col = wave * (16u * TILES_PER_WAVE) + (unsigned)mrow;

#pragma unroll
  for (int tIt = 0; tIt < TILES_PER_WAVE; ++tIt, col += 16u) {
    // B chunk j = rows 4j..4j+3 of X at this column tile (4x16, K x N).
    // Layout: lanes 0-15 {K=4j+0, 4j+1}; lanes 16-31 {K=4j+2, 4j+3}.
    v2f B[4];
#pragma unroll
    for (int j = 0; j < 4; ++j) {
      const size_t r0 = (size_t)(4 * j + 2 * hi);
      B[j].x = __builtin_nontemporal_load(x + (r0    ) * S + col);
      B[j].y = __builtin_nontemporal_load(x + (r0 + 1) * S + col);
    }

    v8f d = {};
#pragma unroll
    for (int j = 0; j < 4; ++j) {
      // 8-arg f32 pattern: (neg_a, A, neg_b, B, c_mod, C, reuse_a, reuse_b)
      d = __builtin_amdgcn_wmma_f32_16x16x4_f32(
              false, A[j], false, B[j], (short)0, d, false, false);
    }

    // D VGPR v: lanes 0-15 -> row v, lanes 16-31 -> row v+8 (N = mrow)
#pragma unroll
    for (int v = 0; v < 8; ++v) {
      __builtin_nontemporal_store(d[v], out + (size_t)(v + 8 * hi) * S + col);
    }
  }
}

extern "C" void kernel_launch(void* const* d_in, const int* in_sizes, int n_in,
                              void* d_out, int out_size, void* d_ws, size_t ws_size,
                              hipStream_t stream) {
  const float* x = (const float*)d_in[0];   // 2^26 f32 state vector
  const float* m = (const float*)d_in[1];   // 4x4 f32 gate, row-major
  float* out = (float*)d_out;               // 2^26 f32

  // 2^22 columns / 64 per wave = 65536 waves; 8 waves per 256-thread block.
  const unsigned total_waves = (1u << ROW_SHIFT) / (16u * TILES_PER_WAVE);
  const unsigned blocks = total_waves / WAVES_PER_BLOCK;   // 8192, exact cover
  hipLaunchKernelGGL(ctrl_gate_wmma, dim3(blocks), dim3(THREADS), 0, stream,
                     x, m, out);
}